// VQCodebook_69329362092038
// MI455X (gfx1250) — compile-verified
//
#include <hip/hip_runtime.h>
#include <hip/hip_bf16.h>

// ---------------------------------------------------------------------------
// VQ codebook quantization for MI455X (gfx1250, wave32).
// Distance cross-term via V_WMMA_F32_16X16X4_F32 (exact fp32 matmul),
// dual independent accumulator chains with sched_barrier-enforced B preload,
// async global->LDS codebook staging (ASYNCcnt-tracked).
// ---------------------------------------------------------------------------

typedef float v2f __attribute__((ext_vector_type(2)));
typedef float v8f __attribute__((ext_vector_type(8)));
typedef int   v4i_vs __attribute__((vector_size(4 * sizeof(int))));  // b128 unit

#define B_DIM 32
#define C_DIM 64
#define W_DIM 64
#define H_DIM 64
#define K_DIM 1024
#define N_DIM (B_DIM * W_DIM * H_DIM)      // 131072 spatial vectors
#define SPATIAL (W_DIM * H_DIM)            // 4096
#define CHW (C_DIM * SPATIAL)              // 262144 floats per batch image

#define CHUNK 256                          // codebook entries staged per LDS pass
#define LDS_PITCH 68                       // 68 % 64 == 4 -> conflict-free B reads
#define LDS_EN_OFF (CHUNK * LDS_PITCH)     // enorm slice after codebook rows
#define BETA 0.25f

#if __has_builtin(__builtin_amdgcn_global_load_async_to_lds_b128)
#define HAVE_ASYNC_LDS 1
#else
#define HAVE_ASYNC_LDS 0
#endif

__device__ __forceinline__ void sched_fence() {
#if __has_builtin(__builtin_amdgcn_sched_barrier)
    __builtin_amdgcn_sched_barrier(0);     // nothing may be moved across
#endif
}

__device__ __forceinline__ void wait_asynccnt0() {
#if __has_builtin(__builtin_amdgcn_s_wait_asynccnt)
    __builtin_amdgcn_s_wait_asynccnt(0);
#else
    asm volatile("s_wait_asynccnt 0x0" ::: "memory");
#endif
}

// --------------------------------------------------------------- stage 0 ----
__global__ __launch_bounds__(256)
void vq_enorm_kernel(const float* __restrict__ emb, float* __restrict__ enorm) {
    int r = blockIdx.x * 256 + threadIdx.x;
    if (r < K_DIM) {
        const float4* p = (const float4*)(emb + r * C_DIM);
        float s = 0.f;
        #pragma unroll
        for (int i = 0; i < C_DIM / 4; ++i) {
            float4 v = p[i];
            s += v.x * v.x + v.y * v.y + v.z * v.z + v.w * v.w;
        }
        enorm[r] = s;
    }
}

// --------------------------------------------------------------- stage 1 ----
// 8 waves / block; each wave owns 16 consecutive z-rows and scans all 1024
// codebook entries with dual chained fp32 WMMA streams.
// dist = ||e||^2 - 2 * (z . e)   (||z||^2 is constant per row -> dropped)
__global__ __launch_bounds__(256)
void vq_argmin_kernel(const float* __restrict__ z,      // [B,C,W,H]
                      const float* __restrict__ emb,    // [K,C]
                      const float* __restrict__ enorm,  // [K]
                      float* __restrict__ idx_f)        // [N] (float-coded ints)
{
    extern __shared__ float lds[];      // CHUNK*LDS_PITCH codebook + CHUNK enorm
    const int tid  = threadIdx.x;
    const int wave = tid >> 5;
    const int lane = tid & 31;
    const int col  = lane & 15;         // M-row for A, N-col for B
    const int hi   = lane >> 4;         // which K-pair of the 4

    const int n0 = blockIdx.x * 128 + wave * 16;        // first of 16 rows
    const int b  = n0 >> 12;                            // n0 / 4096
    const int sp = n0 & 4095;                           // spatial base
    const long zbase = (long)b * CHW + sp + col;        // + c*4096 picks channel

    // A operand: 16 chunks of 16x4 fp32.  lane<16 holds K=(4s,4s+1), lane>=16
    // holds K=(4s+2,4s+3) of row M=col.  z_flat[n][c] = z[b*CHW + c*4096 + sp].
    v2f a[16];
    #pragma unroll
    for (int s = 0; s < 16; ++s) {
        const int c0 = 4 * s + 2 * hi;
        v2f av;
        av.x = z[zbase + (long)c0 * SPATIAL];
        av.y = z[zbase + (long)(c0 + 1) * SPATIAL];
        a[s] = av;
    }

    float minv[8];
    int   mini[8];
    #pragma unroll
    for (int v = 0; v < 8; ++v) { minv[v] = 3.4e38f; mini[v] = 0; }

    for (int chunk = 0; chunk < K_DIM / CHUNK; ++chunk) {
        __syncthreads();   // previous chunk fully consumed
        // ---- stage 256 codebook rows (pitch 68) + their enorms into LDS ----
#if HAVE_ASYNC_LDS
        {
            const float4* gsrc = ((const float4*)emb) + chunk * (CHUNK * C_DIM / 4);
            #pragma unroll
            for (int k = 0; k < (CHUNK * C_DIM / 4) / 256; ++k) {
                const int i4  = tid + k * 256;
                const int row = i4 >> 4;
                const int cl  = (i4 & 15) * 4;
                __builtin_amdgcn_global_load_async_to_lds_b128(
                    (v4i_vs*)(gsrc + i4),
                    (v4i_vs*)&lds[row * LDS_PITCH + cl],
                    0, 0);
            }
        }
        lds[LDS_EN_OFF + tid] = enorm[chunk * CHUNK + tid];
        wait_asynccnt0();
#else
        #pragma unroll
        for (int k = 0; k < (CHUNK * C_DIM / 4) / 256; ++k) {
            const int i4  = tid + k * 256;
            const int row = i4 >> 4;
            const int cl  = (i4 & 15) * 4;
            const float4 v = ((const float4*)emb)[chunk * (CHUNK * C_DIM / 4) + i4];
            float* dst = &lds[row * LDS_PITCH + cl];
            dst[0] = v.x; dst[1] = v.y; dst[2] = v.z; dst[3] = v.w;
        }
        lds[LDS_EN_OFF + tid] = enorm[chunk * CHUNK + tid];
#endif
        __syncthreads();

        // ---- two 16-entry tiles per iteration: independent WMMA chains ----
        #pragma unroll 1
        for (int t = 0; t < CHUNK / 32; ++t) {
            const float* erow0 = &lds[(t * 32 + col) * LDS_PITCH + 2 * hi];
            const float* erow1 = erow0 + 16 * LDS_PITCH;

            // Preload ALL B operands for both tiles, then fence the scheduler
            // so the 16 ds_load_2addr_b64 issue as one clause with a single
            // dscnt wait, instead of being re-sunk to just-in-time position.
            v2f bb0[16], bb1[16];
            #pragma unroll
            for (int s = 0; s < 16; ++s) {
                bb0[s] = *(const v2f*)(erow0 + 4 * s);
                bb1[s] = *(const v2f*)(erow1 + 4 * s);
            }
            sched_fence();

            v8f acc0 = {};
            v8f acc1 = {};
            #pragma unroll
            for (int s = 0; s < 16; ++s) {
                acc0 = __builtin_amdgcn_wmma_f32_16x16x4_f32(
                    false, a[s], false, bb0[s], (short)0, acc0, false, false);
                acc1 = __builtin_amdgcn_wmma_f32_16x16x4_f32(
                    false, a[s], false, bb1[s], (short)0, acc1, false, false);
            }
            sched_fence();

            const int   e0  = chunk * CHUNK + t * 32 + col;
            const float en0 = lds[LDS_EN_OFF + t * 32 + col];
            const float en1 = lds[LDS_EN_OFF + t * 32 + 16 + col];
            #pragma unroll
            for (int v = 0; v < 8; ++v) {
                const float d0 = fmaf(-2.0f, acc0[v], en0);
                const float d1 = fmaf(-2.0f, acc1[v], en1);
                if (d0 < minv[v]) { minv[v] = d0; mini[v] = e0; }
                if (d1 < minv[v]) { minv[v] = d1; mini[v] = e0 + 16; }
            }
        }
    }

    // Reduce (min,idx) across the 16 lanes of each half-wave; ties -> low idx.
    #pragma unroll
    for (int v = 0; v < 8; ++v) {
        float mv = minv[v];
        int   mi = mini[v];
        #pragma unroll
        for (int off = 1; off < 16; off <<= 1) {
            const float ov = __shfl_xor(mv, off, 32);
            const int   oi = __shfl_xor(mi, off, 32);
            if (ov < mv || (ov == mv && oi < mi)) { mv = ov; mi = oi; }
        }
        minv[v] = mv; mini[v] = mi;
    }
    if (col == 0) {
        #pragma unroll
        for (int v = 0; v < 8; ++v)
            idx_f[n0 + hi * 8 + v] = (float)mini[v];    // rows v / v+8
    }
}

// --------------------------------------------------------------- stage 2 ----
__global__ __launch_bounds__(256)
void vq_gather_kernel(const float* __restrict__ z,
                      const float* __restrict__ emb,
                      const float* __restrict__ idx_f,
                      float* __restrict__ quant,        // [B,C,W,H]
                      float* __restrict__ partial)
{
    __shared__ float red[256];
    const int n  = blockIdx.x * 256 + threadIdx.x;
    const int b  = n >> 12;
    const int sp = n & 4095;
    const long zbase = (long)b * CHW + sp;
    const int idx = (int)idx_f[n];
    const float* erow = emb + (long)idx * C_DIM;

    float s = 0.f;
    #pragma unroll
    for (int c = 0; c < C_DIM; ++c) {
        const float q   = erow[c];
        const long  adr = zbase + (long)c * SPATIAL;
        const float d   = q - z[adr];
        s += d * d;
        quant[adr] = q;                                  // coalesced across lanes
    }
    red[threadIdx.x] = s;
    __syncthreads();
    for (int o = 128; o > 0; o >>= 1) {
        if (threadIdx.x < o) red[threadIdx.x] += red[threadIdx.x + o];
        __syncthreads();
    }
    if (threadIdx.x == 0) partial[blockIdx.x] = red[0];
}

// --------------------------------------------------------------- stage 3 ----
__global__ __launch_bounds__(256)
void vq_loss_kernel(const float* __restrict__ partial, float* __restrict__ loss) {
    __shared__ float red[256];
    red[threadIdx.x] = partial[threadIdx.x] + partial[threadIdx.x + 256];
    __syncthreads();
    for (int o = 128; o > 0; o >>= 1) {
        if (threadIdx.x < o) red[threadIdx.x] += red[threadIdx.x + o];
        __syncthreads();
    }
    if (threadIdx.x == 0)
        loss[0] = (1.0f + BETA) * red[0] / (float)(B_DIM * CHW);
}

// ---------------------------------------------------------------------------
extern "C" void kernel_launch(void* const* d_in, const int* in_sizes, int n_in,
                              void* d_out, int out_size, void* d_ws, size_t ws_size,
                              hipStream_t stream) {
    const float* z   = (const float*)d_in[0];   // [32,64,64,64]
    const float* emb = (const float*)d_in[1];   // [1024,64]

    float* out_quant = (float*)d_out;                      // 8388608 floats
    float* out_idx   = out_quant + (size_t)B_DIM * CHW;    // 131072 floats
    float* out_loss  = out_idx + N_DIM;                    // 1 float

    float* ws_enorm   = (float*)d_ws;                      // 1024 floats
    float* ws_partial = ws_enorm + K_DIM;                  // 512 floats

    vq_enorm_kernel<<<(K_DIM + 255) / 256, 256, 0, stream>>>(emb, ws_enorm);

    const size_t lds_bytes =
        (size_t)(CHUNK * LDS_PITCH + CHUNK) * sizeof(float);  // 70656 B
    vq_argmin_kernel<<<N_DIM / 128, 256, lds_bytes, stream>>>(
        z, emb, ws_enorm, out_idx);

    vq_gather_kernel<<<N_DIM / 256, 256, 0, stream>>>(
        z, emb, out_idx, out_quant, ws_partial);

    vq_loss_kernel<<<1, 256, 0, stream>>>(ws_partial, out_loss);
}